// SimpleSSM_81827716923495
// MI455X (gfx1250) — compile-verified
//
#include <hip/hip_runtime.h>
#include <hip/hip_bf16.h>

typedef __attribute__((ext_vector_type(2))) float v2f;
typedef __attribute__((ext_vector_type(4))) float v4f;
typedef __attribute__((ext_vector_type(8))) float v8f;
typedef __attribute__((ext_vector_type(4))) int   v4i;

// D = A(16x4) * B(4x16) + C(16x16), all f32, wave32.
#define WMMA_F32(a, b, c)                                                     \
  __builtin_amdgcn_wmma_f32_16x16x4_f32(false, (a), false, (b), (short)0,     \
                                        (c), false, false)

#ifndef __has_builtin
#define __has_builtin(x) 0
#endif
#if __has_builtin(__builtin_amdgcn_global_load_async_to_lds_b128)
#define HAVE_ASYNC_LDS 1
#else
#define HAVE_ASYNC_LDS 0
#endif

#define GAS __attribute__((address_space(1)))
#define LAS __attribute__((address_space(3)))

#if HAVE_ASYNC_LDS
// param types from clang diagnostic: (AS1 v4i*, AS3 v4i*, imm off, imm cpol)
#define ASYNC_CP16(lds_dst, glb_src)                                          \
  __builtin_amdgcn_global_load_async_to_lds_b128((GAS v4i*)(glb_src),         \
                                                 (LAS v4i*)(lds_dst), 0, 0)
#endif

__device__ __forceinline__ void wait_asynccnt0() {
#if __has_builtin(__builtin_amdgcn_s_wait_asynccnt)
  __builtin_amdgcn_s_wait_asynccnt(0);
#else
  asm volatile("s_wait_asynccnt 0x0" ::: "memory");
#endif
}

// Opaque pointer copy: defeats LICM so invariant global fragment loads are
// re-issued per scan step instead of being hoisted into ~128 live VGPRs
// (round-1 disasm showed scratch spill/reload from exactly that hoisting).
// Applied to an addrspace(1) pointer so the loads lower to global_load_b64
// (round-3 disasm showed flat_load_b64, which also ticks DScnt and contends
// with the real LDS fragment traffic).
#define ANTI_HOIST(p) asm volatile("" : "+v"(p))

#define BATCH 16
#define SEQ 4096
#define DH 256
#define NCLS 3
#define LCH 64
#define NCHUNK (SEQ / LCH)
#define HSTRIDE 260  // padded LDS row stride (floats): 4-bank shift per row

// ---------------------------------------------------------------------------
// Kernel 1: per-chunk local scan; xB projection as a separate WMMA chain, added to h@A^T.
// 16 waves per chunk; wave w owns hidden columns [16w,16w+16).
//   cu = x_t @ B^T   (independent chain, B^T streamed from L2 per step)
//   ch = h   @ A^T   (serial chain, A^T frags resident in 128 VGPRs)
//   h' = cu + ch     (published through padded LDS)
// x_t staged into double-buffered LDS with async-to-LDS loads (ASYNCcnt).
// ---------------------------------------------------------------------------
__global__ __launch_bounds__(512, 1)
void ssm_chunk_scan(const float* __restrict__ x,   // (16, 4096, 256)
                    const float* __restrict__ Bm,  // (256, 256)
                    const float* __restrict__ Am,  // (256, 256)
                    float* __restrict__ hfin,      // (NCHUNK, 16, 256)
                    float* __restrict__ accl) {    // (NCHUNK, 16, 256)
  __shared__ float hsm[16 * HSTRIDE];
  __shared__ float xsm[2][16 * HSTRIDE];

  const int chunk = blockIdx.x;
  const int lane  = threadIdx.x & 31;
  const int wave  = threadIdx.x >> 5;
  const int lrow  = lane & 15;
  const int hi    = lane >> 4;
  const int khalf = hi * 2;

  for (int i = threadIdx.x; i < 16 * HSTRIDE; i += 512) hsm[i] = 0.f;

  // A^T fragments resident for the whole chunk: At[kb] = Am[16w+n][4kb+k]
  v2f atf[64];
  {
    const float* arow = Am + (16 * wave + lrow) * DH + khalf;
#pragma unroll
    for (int kb = 0; kb < 64; ++kb) atf[kb] = *(const v2f*)(arow + 4 * kb);
  }
  const GAS float* brow = (const GAS float*)(Bm + (16 * wave + lrow) * DH + khalf);

  // cooperative x_t staging: thread -> 8 consecutive d of one batch row
  const int cb  = threadIdx.x >> 5;
  const int cd0 = (threadIdx.x & 31) * 8;
  const float* xg = x + (size_t)cb * (SEQ * DH) + cd0;
  {
    const int t0 = chunk * LCH;
#if HAVE_ASYNC_LDS
    ASYNC_CP16(&xsm[0][cb * HSTRIDE + cd0],     xg + (size_t)t0 * DH);
    ASYNC_CP16(&xsm[0][cb * HSTRIDE + cd0 + 4], xg + (size_t)t0 * DH + 4);
    wait_asynccnt0();
#else
    *(v4f*)&xsm[0][cb * HSTRIDE + cd0]     = *(const v4f*)(xg + (size_t)t0 * DH);
    *(v4f*)&xsm[0][cb * HSTRIDE + cd0 + 4] = *(const v4f*)(xg + (size_t)t0 * DH + 4);
#endif
  }
  __syncthreads();

  v8f acc   = {};
  v8f hlast = {};
  int cur = 0;
  for (int i = 0; i < LCH; ++i) {
    const int t  = chunk * LCH + i;
    const int tn = (t + 1 < SEQ) ? t + 1 : t;  // clamped; value unused at end

    // stage x_{t+1} into the other buffer (safe: all waves finished reading
    // it at the end of the previous step, past the second barrier)
#if HAVE_ASYNC_LDS
    ASYNC_CP16(&xsm[cur ^ 1][cb * HSTRIDE + cd0],     xg + (size_t)tn * DH);
    ASYNC_CP16(&xsm[cur ^ 1][cb * HSTRIDE + cd0 + 4], xg + (size_t)tn * DH + 4);
#else
    v4f nx0 = *(const v4f*)(xg + (size_t)tn * DH);
    v4f nx1 = *(const v4f*)(xg + (size_t)tn * DH + 4);
#endif

    const GAS float* bstep = brow;
    ANTI_HOIST(bstep);  // re-stream B^T frags from cache each step (no hoist)

    // independent chain: cu = x_t @ B^T
    v8f cu = {};
#pragma unroll
    for (int kb = 0; kb < 64; ++kb) {
      v2f a = *(const v2f*)(&xsm[cur][lrow * HSTRIDE + 4 * kb + khalf]);
      v2f b = *(const GAS v2f*)(bstep + 4 * kb);
      cu = WMMA_F32(a, b, cu);
    }
    // serial chain: ch = h @ A^T (A^T from registers)
    v8f ch = {};
#pragma unroll
    for (int kb = 0; kb < 64; ++kb) {
      v2f a = *(const v2f*)(&hsm[lrow * HSTRIDE + 4 * kb + khalf]);
      ch = WMMA_F32(a, atf[kb], ch);
    }
    v8f c;
#pragma unroll
    for (int r = 0; r < 8; ++r) c[r] = cu[r] + ch[r];

    __syncthreads();  // everyone done reading old h
#pragma unroll
    for (int r = 0; r < 8; ++r) {
      hsm[(r + 8 * hi) * HSTRIDE + 16 * wave + lrow] = c[r];
      acc[r] += c[r];
    }
#if HAVE_ASYNC_LDS
    wait_asynccnt0();  // x_{t+1} landed in LDS before the release barrier
#else
    *(v4f*)&xsm[cur ^ 1][cb * HSTRIDE + cd0]     = nx0;
    *(v4f*)&xsm[cur ^ 1][cb * HSTRIDE + cd0 + 4] = nx1;
#endif
    hlast = c;
    cur ^= 1;
    __syncthreads();
  }

  float* hf = hfin + (size_t)chunk * BATCH * DH;
  float* al = accl + (size_t)chunk * BATCH * DH;
#pragma unroll
  for (int r = 0; r < 8; ++r) {
    const int m = r + 8 * hi;
    hf[m * DH + 16 * wave + lrow] = hlast[r];
    al[m * DH + 16 * wave + lrow] = acc[r];
  }
}

// ---------------------------------------------------------------------------
// Kernel 2: 256x256x256 f32 WMMA matmul  D = X @ Y (+ Cin), for log-doubling.
// ---------------------------------------------------------------------------
__global__ __launch_bounds__(256, 1)
void mm256(const float* __restrict__ X, const float* __restrict__ Y,
           const float* __restrict__ Cin, float* __restrict__ D) {
  const int lane  = threadIdx.x & 31;
  const int tile  = blockIdx.x * 8 + (threadIdx.x >> 5);
  const int mt    = tile >> 4, nt = tile & 15;
  const int lrow  = lane & 15;
  const int hi    = lane >> 4;
  const int khalf = hi * 2;

  v8f c = {};
  if (Cin) {
#pragma unroll
    for (int r = 0; r < 8; ++r)
      c[r] = Cin[(16 * mt + r + 8 * hi) * DH + 16 * nt + lrow];
  }
  const float* xr = X + (16 * mt + lrow) * DH + khalf;
#pragma unroll
  for (int kb = 0; kb < 64; ++kb) {
    v2f a = *(const v2f*)(xr + 4 * kb);
    const int k0 = 4 * kb + khalf;
    v2f b;
    b[0] = Y[(k0    ) * DH + 16 * nt + lrow];
    b[1] = Y[(k0 + 1) * DH + 16 * nt + lrow];
    c = WMMA_F32(a, b, c);
  }
#pragma unroll
  for (int r = 0; r < 8; ++r)
    D[(16 * mt + r + 8 * hi) * DH + 16 * nt + lrow] = c[r];
}

// ---------------------------------------------------------------------------
// Kernel 3: sequential combine over chunks + pooling + head.
//   g' = g @ M1^T + hfin_c ;  acc += g @ M2^T + accl_c
// M1^T frags resident in registers (serial chain); M2^T streamed per chunk.
// ---------------------------------------------------------------------------
__global__ __launch_bounds__(512, 1)
void ssm_combine(const float* __restrict__ hfin, const float* __restrict__ accl,
                 const float* __restrict__ M1, const float* __restrict__ M2,
                 const float* __restrict__ Cm, const float* __restrict__ Wh,
                 const float* __restrict__ bh, float* __restrict__ out) {
  __shared__ float gsm[16 * HSTRIDE];
  __shared__ float ysm[16 * DH];

  const int lane  = threadIdx.x & 31;
  const int wave  = threadIdx.x >> 5;
  const int lrow  = lane & 15;
  const int hi    = lane >> 4;
  const int khalf = hi * 2;

  for (int i = threadIdx.x; i < 16 * HSTRIDE; i += 512) gsm[i] = 0.f;

  v2f m1f[64];
  {
    const float* p = M1 + (16 * wave + lrow) * DH + khalf;
#pragma unroll
    for (int kb = 0; kb < 64; ++kb) m1f[kb] = *(const v2f*)(p + 4 * kb);
  }
  const GAS float* m2row =
      (const GAS float*)(M2 + (16 * wave + lrow) * DH + khalf);

  v8f accT = {};
  __syncthreads();

  for (int ck = 0; ck < NCHUNK; ++ck) {
    const float* hf = hfin + (size_t)ck * BATCH * DH;
    const float* al = accl + (size_t)ck * BATCH * DH;
    const GAS float* m2p = m2row;
    ANTI_HOIST(m2p);
    v8f cg = {}, ca = {};
#pragma unroll
    for (int r = 0; r < 8; ++r) {
      const int m = r + 8 * hi;
      cg[r] = hf[m * DH + 16 * wave + lrow];
      ca[r] = al[m * DH + 16 * wave + lrow];
    }
#pragma unroll
    for (int kb = 0; kb < 64; ++kb) {
      v2f a  = *(const v2f*)(&gsm[lrow * HSTRIDE + 4 * kb + khalf]);
      v2f b2 = *(const GAS v2f*)(m2p + 4 * kb);
      cg = WMMA_F32(a, m1f[kb], cg);  // two independent chains
      ca = WMMA_F32(a, b2, ca);
    }
#pragma unroll
    for (int r = 0; r < 8; ++r) accT[r] += ca[r];
    __syncthreads();
#pragma unroll
    for (int r = 0; r < 8; ++r)
      gsm[(r + 8 * hi) * HSTRIDE + 16 * wave + lrow] = cg[r];
    __syncthreads();
  }

  // pooled = accT / SEQ  -> reuse gsm as A-operand buffer
#pragma unroll
  for (int r = 0; r < 8; ++r)
    gsm[(r + 8 * hi) * HSTRIDE + 16 * wave + lrow] = accT[r] * (1.0f / SEQ);
  __syncthreads();

  // y = pooled @ C^T
  const float* crow = Cm + (16 * wave + lrow) * DH + khalf;
  v8f cy = {};
#pragma unroll
  for (int kb = 0; kb < 64; ++kb) {
    v2f a = *(const v2f*)(&gsm[lrow * HSTRIDE + 4 * kb + khalf]);
    v2f b = *(const v2f*)(crow + 4 * kb);
    cy = WMMA_F32(a, b, cy);
  }
#pragma unroll
  for (int r = 0; r < 8; ++r)
    ysm[(r + 8 * hi) * DH + 16 * wave + lrow] = cy[r];
  __syncthreads();

  if (threadIdx.x < BATCH * NCLS) {
    const int b = threadIdx.x / NCLS, k = threadIdx.x % NCLS;
    float s = bh[k];
    for (int o = 0; o < DH; ++o) s += ysm[b * DH + o] * Wh[k * DH + o];
    out[b * NCLS + k] = s;
  }
}

extern "C" void kernel_launch(void* const* d_in, const int* in_sizes, int n_in,
                              void* d_out, int out_size, void* d_ws,
                              size_t ws_size, hipStream_t stream) {
  const float* x  = (const float*)d_in[0];
  const float* A  = (const float*)d_in[1];
  const float* Bm = (const float*)d_in[2];
  const float* Cm = (const float*)d_in[3];
  const float* Wh = (const float*)d_in[4];
  const float* bh = (const float*)d_in[5];
  float* out = (float*)d_out;

  char* ws = (char*)d_ws;
  const size_t MATB = (size_t)DH * DH * sizeof(float);  // 256 KB
  float* Pb[2] = {(float*)(ws), (float*)(ws + MATB)};
  float* Sb[2] = {(float*)(ws + 2 * MATB), (float*)(ws + 3 * MATB)};
  float* hfin  = (float*)(ws + 4 * MATB);
  float* accl  = (float*)(ws + 4 * MATB + (size_t)NCHUNK * BATCH * DH * 4);
  // total ws use: 3 MB

  ssm_chunk_scan<<<NCHUNK, 512, 0, stream>>>(x, Bm, A, hfin, accl);

  // M1 = A^64, M2 = sum_{j=1..64} A^j via log-doubling:
  //   S_{2n} = P_n @ S_n + S_n ;  P_{2n} = P_n @ P_n ; start P_1 = S_1 = A
  (void)hipMemcpyAsync(Pb[0], A, MATB, hipMemcpyDeviceToDevice, stream);
  (void)hipMemcpyAsync(Sb[0], A, MATB, hipMemcpyDeviceToDevice, stream);
  for (int d = 0; d < 6; ++d) {
    const int s = d & 1, t = s ^ 1;
    mm256<<<32, 256, 0, stream>>>(Pb[s], Sb[s], Sb[s], Sb[t]);
    mm256<<<32, 256, 0, stream>>>(Pb[s], Pb[s], nullptr, Pb[t]);
  }

  ssm_combine<<<1, 512, 0, stream>>>(hfin, accl, Pb[0], Sb[0], Cm, Wh, bh, out);
}